// NestedRecursiveLogitRouteChoice_63488206570158
// MI455X (gfx1250) — compile-verified
//
#include <hip/hip_runtime.h>
#include <hip/hip_bf16.h>
#include <math.h>

typedef float v2f __attribute__((ext_vector_type(2)));
typedef float v8f __attribute__((ext_vector_type(8)));

#define EPSV 1e-30f
#define DFEAT 64
#define NITERS 40

// ---------------------------------------------------------------------------
// Kernel 1: edge encoder GEMV via f32 WMMA.
// Each wave handles 16 edges (rows). A-frag (16x4 f32): lane L (0-15) holds
// row L, K = {4k, 4k+1}; lanes 16-31 hold row L-16, K = {4k+2, 4k+3}.
// B-frag: W chunk replicated across all 16 columns, so D[m][n] is identical
// for every n. Per D layout, lane 0 holds edges 0..7 (VGPRs 0..7) and lane 16
// holds edges 8..15 — those two lanes store the raw logits.
// ---------------------------------------------------------------------------
__global__ void k_gemv_wmma(const float* __restrict__ feats,
                            const float* __restrict__ W,
                            const float* __restrict__ bptr,
                            float* __restrict__ sraw, int E) {
  const int lane = threadIdx.x & 31;
  const long wave = (long)blockIdx.x * (blockDim.x >> 5) + (threadIdx.x >> 5);
  const long base = wave * 16;
  if (base >= E) return;                  // wave-uniform: EXEC stays full
  const int hi  = lane >> 4;              // 0: lanes 0-15, 1: lanes 16-31
  const int row = lane & 15;
  long erow = base + row;
  if (erow >= E) erow = (long)E - 1;      // clamp (tail safety; loads only)
  const float* rp = feats + erow * DFEAT + (hi << 1);
  const float* wp = W + (hi << 1);

  v8f c = {};
#pragma unroll
  for (int k = 0; k < DFEAT / 4; ++k) {
    v2f a = *(const v2f*)(rp + 4 * k);    // 8B-aligned pair of K values
    v2f b = *(const v2f*)(wp + 4 * k);    // broadcast load of W chunk
    c = __builtin_amdgcn_wmma_f32_16x16x4_f32(
        /*neg_a=*/false, a, /*neg_b=*/false, b,
        /*c_mod=*/(short)0, c, /*reuse_a=*/false, /*reuse_b=*/false);
  }

  const float bias = bptr[0];
  if (row == 0) {                         // lanes 0 and 16 only
    const long ebase = base + (hi << 3);
#pragma unroll
    for (int m = 0; m < 8; ++m) {
      const long e = ebase + m;
      if (e < E) sraw[e] = c[m] + bias;
    }
  }
}

// ---------------------------------------------------------------------------
// Kernel 2: per-edge prep. rewards = -softplus(s); M = exp(r/mu_i); p = mu_j/mu_i
// ---------------------------------------------------------------------------
__global__ void k_edgeprep(const float* __restrict__ sraw,
                           const float* __restrict__ scales,
                           const int* __restrict__ src,
                           const int* __restrict__ dst,
                           float* __restrict__ rewards,
                           float* __restrict__ M,
                           float* __restrict__ p, int E) {
  const int e = blockIdx.x * blockDim.x + threadIdx.x;
  if (e >= E) return;
  const float z  = sraw[e];
  const float sp = (z > 0.f) ? (z + log1pf(expf(-z))) : log1pf(expf(z));
  const float r  = -sp;
  rewards[e] = r;
  const float mi = scales[dst[e]];
  const float mj = scales[src[e]];
  M[e] = expf(r / mi);
  p[e] = mj / mi;
}

// x0 = sink
__global__ void k_init(const float* __restrict__ sink, float* __restrict__ x, int N) {
  const int i = blockIdx.x * blockDim.x + threadIdx.x;
  if (i < N) x[i] = sink[i];
}

// per-iteration node pass: x_next = sink; lx = log(x_cur)  (log(0) = -inf -> exp gives 0)
__global__ void k_prepiter(const float* __restrict__ xc,
                           const float* __restrict__ sink,
                           float* __restrict__ xn,
                           float* __restrict__ lx, int N) {
  const int i = blockIdx.x * blockDim.x + threadIdx.x;
  if (i >= N) return;
  xn[i] = sink[i];
  lx[i] = logf(xc[i]);
}

// per-iteration edge pass: x_next[dst] += M * exp(p * lx[src])
// unsafeAtomicAdd forces the hardware global_atomic_add_f32 (no CAS loop);
// addends here are exp()-products in normal range, so denorm-flush semantics
// are irrelevant.
__global__ void k_scatter(const int* __restrict__ src,
                          const int* __restrict__ dst,
                          const float* __restrict__ M,
                          const float* __restrict__ p,
                          const float* __restrict__ lx,
                          float* __restrict__ xn, int E) {
  const int e = blockIdx.x * blockDim.x + threadIdx.x;
  if (e >= E) return;
  const float z = expf(p[e] * lx[src[e]]);
  unsafeAtomicAdd(&xn[dst[e]], M[e] * z);
}

__global__ void k_values(const float* __restrict__ x, float* __restrict__ vals, int N) {
  const int i = blockIdx.x * blockDim.x + threadIdx.x;
  if (i < N) vals[i] = logf(fmaxf(x[i], EPSV));
}

__global__ void k_eprobs(const int* __restrict__ src,
                         const int* __restrict__ dst,
                         const float* __restrict__ M,
                         const float* __restrict__ p,
                         const float* __restrict__ x,
                         float* __restrict__ eprobs, int E) {
  const int e = blockIdx.x * blockDim.x + threadIdx.x;
  if (e >= E) return;
  const float msg   = M[e] * powf(x[src[e]], p[e]);
  const float denom = x[dst[e]];
  eprobs[e] = (denom > 0.f) ? (msg / fmaxf(denom, EPSV)) : 0.f;
}

// ---------------------------------------------------------------------------
extern "C" void kernel_launch(void* const* d_in, const int* in_sizes, int n_in,
                              void* d_out, int out_size, void* d_ws, size_t ws_size,
                              hipStream_t stream) {
  (void)n_in; (void)out_size; (void)ws_size;
  const int*   edge_index = (const int*)d_in[0];
  const float* edge_feats = (const float*)d_in[1];
  const float* scales     = (const float*)d_in[2];
  const float* sink       = (const float*)d_in[3];
  const float* W          = (const float*)d_in[4];
  const float* b          = (const float*)d_in[5];

  const int E = in_sizes[0] / 2;
  const int N = in_sizes[2];
  const int* src = edge_index;
  const int* dst = edge_index + E;

  float* out     = (float*)d_out;
  float* rewards = out;
  float* values  = out + E;
  float* eprobs  = out + (size_t)E + N;

  float* ws = (float*)d_ws;
  float* sraw = ws;
  float* M    = ws + (size_t)E;
  float* p    = ws + 2 * (size_t)E;
  float* x0   = ws + 3 * (size_t)E;
  float* x1   = x0 + N;
  float* lx   = x1 + N;

  const dim3 blk(256);
  const int wavesNeeded = (E + 15) / 16;
  const dim3 gW((wavesNeeded + 7) / 8);          // 8 waves / block
  const dim3 gE((E + 255) / 256);
  const dim3 gN((N + 255) / 256);

  k_gemv_wmma<<<gW, blk, 0, stream>>>(edge_feats, W, b, sraw, E);
  k_edgeprep<<<gE, blk, 0, stream>>>(sraw, scales, src, dst, rewards, M, p, E);
  k_init<<<gN, blk, 0, stream>>>(sink, x0, N);

  float* xc = x0;
  float* xn = x1;
  for (int t = 0; t < NITERS; ++t) {
    k_prepiter<<<gN, blk, 0, stream>>>(xc, sink, xn, lx, N);
    k_scatter<<<gE, blk, 0, stream>>>(src, dst, M, p, lx, xn, E);
    float* tmp = xc; xc = xn; xn = tmp;
  }

  k_values<<<gN, blk, 0, stream>>>(xc, values, N);
  k_eprobs<<<gE, blk, 0, stream>>>(src, dst, M, p, xc, eprobs, E);
}